// WiredRNN_45260365365275
// MI455X (gfx1250) — compile-verified
//
#include <hip/hip_runtime.h>

#define N_IN  512
#define N_OUT 256
#define BATCH 32

#define CHUNK   1024   // edges staged per TDM chunk (12 KB per buffer); mult of 32
#define NWAVES  8      // 256 threads / wave32

typedef __attribute__((ext_vector_type(4))) unsigned int u32x4;
typedef __attribute__((ext_vector_type(8))) int          i32x8;
typedef __attribute__((ext_vector_type(4))) int          i32x4;

static __device__ __forceinline__ long lmin(long a, long b) { return a < b ? a : b; }

// ---------------------------------------------------------------------------
// TDM: 1-D async copy of n 4-byte elements global -> LDS.
// D# packed per CDNA5 ISA §8.3/8.4: group0 = {count=1, lds_addr,
// global_addr[56:0], type=2}; group1 = {data_size=4B, tensor_dim0=n,
// tensor_dim1=1, tile_dim0=n, tile_dim1/2=0, stride0=n}. Groups 2/3 zero
// (<=2D tensor). Tracked by TENSORcnt.
// ---------------------------------------------------------------------------
static __device__ __forceinline__ void tdm_load_1d(const void* gptr, void* lptr,
                                                   unsigned n) {
  unsigned long long ga = (unsigned long long)(size_t)gptr;
  unsigned lds = (unsigned)(size_t)lptr;  // low 32 bits of generic ptr = LDS offset
  u32x4 g0;
  g0.x = 1u;                                                // count=1, user mode
  g0.y = lds;                                               // lds_addr (bytes)
  g0.z = (unsigned)(ga & 0xFFFFFFFFu);                      // global_addr[31:0]
  g0.w = (unsigned)((ga >> 32) & 0x01FFFFFFu) | (2u << 30); // ga[56:32] | type=2
  i32x8 g1;
  g1[0] = 0x00020000;                                       // data_size=2 (4B)
  g1[1] = (int)((n & 0xFFFFu) << 16);                       // tensor_dim0 lo16
  g1[2] = (int)((n >> 16) & 0xFFFFu) | (1 << 16);           // dim0 hi16 | dim1=1
  g1[3] = (int)((n & 0xFFFFu) << 16);                       // tile_dim0 = n
  g1[4] = 0;                                                // tile_dim1/2 = 0 (1-D)
  g1[5] = (int)n;                                           // dim0_stride = n
  g1[6] = 0;
  g1[7] = 0;
  i32x4 z4 = {0, 0, 0, 0};
#if defined(__clang_major__) && (__clang_major__ >= 23)
  i32x8 z8 = {0, 0, 0, 0, 0, 0, 0, 0};
  __builtin_amdgcn_tensor_load_to_lds(g0, g1, z4, z4, z8, 0);
#else
  __builtin_amdgcn_tensor_load_to_lds(g0, g1, z4, z4, 0);
#endif
}

// ---------------------------------------------------------------------------
// Kernel A: states (B,N) -> states_T (N,B) via LDS 32x32 tile transpose,
// and zero agg_T (N,B). Both sides fully coalesced.
// ---------------------------------------------------------------------------
__global__ __launch_bounds__(256) void prep_kernel(
    const float* __restrict__ states, float* __restrict__ states_T,
    float* __restrict__ agg_T, int N) {
  __shared__ float tile[32][33];  // [batch][node_local], pad avoids conflicts
  const int n0 = blockIdx.x * 32;
  const int tx = threadIdx.x;  // 0..31
  const int ty = threadIdx.y;  // 0..7
#pragma unroll
  for (int i = 0; i < 32; i += 8) {
    const int b = ty + i;
    const int n = n0 + tx;
    tile[b][tx] = (n < N) ? states[(size_t)b * N + n] : 0.0f;  // coalesced over n
  }
  __syncthreads();
#pragma unroll
  for (int i = 0; i < 32; i += 8) {
    const int n = n0 + ty + i;
    if (n < N) {
      states_T[(size_t)n * BATCH + tx] = tile[tx][ty + i];  // coalesced over b
      agg_T[(size_t)n * BATCH + tx] = 0.0f;
    }
  }
}

// ---------------------------------------------------------------------------
// Kernel B: edge scatter. lane = batch (wave32 == B=32). Edge metadata is
// double-buffered into LDS by the TDM (wave 0 issues, TENSORcnt ordering);
// processing of chunk c overlaps the DMA of chunk c+1. Inner loop handles
// 4 edges per iteration: one ds_load_b128 per array, then 4 independent 128B
// gathers (MLP) followed by 4 no-return global_atomic_add_f32 into
// L2-resident agg_T.
// ---------------------------------------------------------------------------
__global__ __launch_bounds__(256) void edge_kernel(
    const int* __restrict__ edge_src, const int* __restrict__ edge_dst,
    const float* __restrict__ weight, const float* __restrict__ states_T,
    float* __restrict__ agg_T, int E, int edges_per_block) {
  __shared__ __align__(16) int   s_src[2][CHUNK];
  __shared__ __align__(16) int   s_dst[2][CHUNK];
  __shared__ __align__(16) float s_w[2][CHUNK];

  const int lane = (int)(threadIdx.x & 31u);
  const int wave = __builtin_amdgcn_readfirstlane((int)(threadIdx.x >> 5));

  const long e_begin = (long)blockIdx.x * (long)edges_per_block;
  const long e_end   = lmin((long)E, e_begin + (long)edges_per_block);
  if (e_begin >= e_end) return;
  const long nchunks = (e_end - e_begin + CHUNK - 1) / CHUNK;

  if (wave == 0) {
    const unsigned n0 = (unsigned)lmin((long)CHUNK, e_end - e_begin);
    tdm_load_1d(edge_src + e_begin, &s_src[0][0], n0);
    tdm_load_1d(edge_dst + e_begin, &s_dst[0][0], n0);
    tdm_load_1d(weight   + e_begin, &s_w[0][0],   n0);
  }

  for (long c = 0; c < nchunks; ++c) {
    const int buf = (int)(c & 1);
    if (wave == 0) {
      if (c + 1 < nchunks) {  // prefetch next chunk into the other buffer
        const long base = e_begin + (c + 1) * CHUNK;
        const unsigned nn = (unsigned)lmin((long)CHUNK, e_end - base);
        const int nb = buf ^ 1;
        tdm_load_1d(edge_src + base, &s_src[nb][0], nn);
        tdm_load_1d(edge_dst + base, &s_dst[nb][0], nn);
        tdm_load_1d(weight   + base, &s_w[nb][0],   nn);
        __builtin_amdgcn_s_wait_tensorcnt(3);  // chunk c done (in-order per wave)
      } else {
        __builtin_amdgcn_s_wait_tensorcnt(0);
      }
    }
    __syncthreads();  // chunk c visible in LDS to all waves

    const int nelem = (int)lmin((long)CHUNK, e_end - (e_begin + c * CHUNK));

    // Main path: 4 edges per iteration (full chunks are multiples of 32).
    const int ngroups = nelem >> 2;
    for (int g = wave; g < ngroups; g += NWAVES) {
      const int4   s4 = *(const int4*)(&s_src[buf][g << 2]);   // ds_load_b128
      const int4   d4 = *(const int4*)(&s_dst[buf][g << 2]);
      const float4 w4 = *(const float4*)(&s_w[buf][g << 2]);
      // 4 independent 128B gathers in flight before the atomics start
      const float v0 = states_T[(size_t)s4.x * BATCH + lane] * w4.x;
      const float v1 = states_T[(size_t)s4.y * BATCH + lane] * w4.y;
      const float v2 = states_T[(size_t)s4.z * BATCH + lane] * w4.z;
      const float v3 = states_T[(size_t)s4.w * BATCH + lane] * w4.w;
      unsafeAtomicAdd(&agg_T[(size_t)d4.x * BATCH + lane], v0);
      unsafeAtomicAdd(&agg_T[(size_t)d4.y * BATCH + lane], v1);
      unsafeAtomicAdd(&agg_T[(size_t)d4.z * BATCH + lane], v2);
      unsafeAtomicAdd(&agg_T[(size_t)d4.w * BATCH + lane], v3);
    }
    // Tail (only the final partial chunk of the block reaches here).
    for (int i = (ngroups << 2) + wave; i < nelem; i += NWAVES) {
      const int   src = s_src[buf][i];
      const int   dst = s_dst[buf][i];
      const float w   = s_w[buf][i];
      const float v = states_T[(size_t)src * BATCH + lane] * w;
      unsafeAtomicAdd(&agg_T[(size_t)dst * BATCH + lane], v);
    }
    __syncthreads();  // everyone done with buf before TDM reuses it
  }
}

// ---------------------------------------------------------------------------
// Kernel C: z = tanh(bias + response*agg), clamp input nodes to x, emit
// new_states (B,N) and y = last N_OUT columns. LDS tile transpose so agg_T
// reads and new_states writes are both coalesced; bias/response are
// wave-uniform -> scalar loads.
// ---------------------------------------------------------------------------
__global__ __launch_bounds__(256) void final_kernel(
    const float* __restrict__ agg_T, const float* __restrict__ bias,
    const float* __restrict__ response, const float* __restrict__ x,
    float* __restrict__ out_y, float* __restrict__ out_states, int N) {
  __shared__ float tile[32][33];  // [node_local][batch]
  const int n0 = blockIdx.x * 32;
  const int tx = threadIdx.x;  // 0..31 (batch in phase 1, node in phase 2)
  const int ty = threadIdx.y;  // 0..7
#pragma unroll
  for (int i = 0; i < 32; i += 8) {
    const int nl = ty + i;
    const int n = n0 + nl;
    float z = 0.0f;
    if (n < N) {
      const float a = agg_T[(size_t)n * BATCH + tx];  // coalesced over batch
      z = tanhf(bias[n] + response[n] * a);           // bias/response scalar
      if (n < N_IN) z = x[tx * N_IN + n];             // clamp input nodes
    }
    tile[nl][tx] = z;
  }
  __syncthreads();
#pragma unroll
  for (int i = 0; i < 32; i += 8) {
    const int b = ty + i;
    const int n = n0 + tx;
    if (n < N) {
      const float z = tile[tx][b];  // stride-33 -> conflict-free
      out_states[(size_t)b * N + n] = z;  // coalesced over n
      if (n >= N - N_OUT) out_y[b * N_OUT + (n - (N - N_OUT))] = z;
    }
  }
}

extern "C" void kernel_launch(void* const* d_in, const int* in_sizes, int n_in,
                              void* d_out, int out_size, void* d_ws, size_t ws_size,
                              hipStream_t stream) {
  const float* x        = (const float*)d_in[0];  // (32, 512)
  const float* states   = (const float*)d_in[1];  // (32, N)
  const float* weight   = (const float*)d_in[2];  // (E,)
  const float* bias     = (const float*)d_in[3];  // (N,)
  const float* response = (const float*)d_in[4];  // (N,)
  const int*   edge_src = (const int*)d_in[5];    // (E,)
  const int*   edge_dst = (const int*)d_in[6];    // (E,)

  const int N = in_sizes[1] / BATCH;  // 50000
  const int E = in_sizes[2];          // 800000

  float* states_T = (float*)d_ws;                       // N*32 floats (6.4 MB)
  float* agg_T    = states_T + (size_t)N * BATCH;       // N*32 floats (6.4 MB)

  float* out_y      = (float*)d_out;                    // (32, 256)
  float* out_states = (float*)d_out + BATCH * N_OUT;    // (32, N)

  const int nTiles = (N + 31) / 32;
  dim3 tb(32, 8);

  prep_kernel<<<nTiles, tb, 0, stream>>>(states, states_T, agg_T, N);

  const int nblocks = 256;  // 3125 edges/block -> 4 TDM chunks, 2048 waves
  const int epb = (E + nblocks - 1) / nblocks;
  edge_kernel<<<nblocks, 256, 0, stream>>>(edge_src, edge_dst, weight,
                                           states_T, agg_T, E, epb);

  final_kernel<<<nTiles, tb, 0, stream>>>(agg_T, bias, response, x,
                                          out_y, out_states, N);
}